// GIN_60842506715387
// MI455X (gfx1250) — compile-verified
//
#include <hip/hip_runtime.h>

#define D        128
#define NNODES   50000
#define NEDGES   600000
#define NGRAPHS  512
#define NLAYERS  4
#define NCLASSES 10
#define ROWS_PER_BLOCK 64

typedef __attribute__((ext_vector_type(2))) float v2f;
typedef __attribute__((ext_vector_type(8))) float v8f;

// ---------------------------------------------------------------- utilities
__global__ void zero_f32(float* __restrict__ p, int n) {
    int i = blockIdx.x * blockDim.x + threadIdx.x;
    int stride = gridDim.x * blockDim.x;
    for (; i < n; i += stride) p[i] = 0.0f;
}

// ------------------------------------------------- edge scatter (sum aggregate)
// 2 edges per 256-thread block; 128 lanes move one full feature row per edge.
__global__ __launch_bounds__(256) void scatter_add_edges(
    const float* __restrict__ h, const long long* __restrict__ src,
    const long long* __restrict__ dst, float* __restrict__ agg) {
    int e = blockIdx.x * 2 + (threadIdx.x >> 7);
    int t = threadIdx.x & 127;
    if (e < NEDGES) {
        long long s = src[e];
        long long d = dst[e];
        atomicAdd(&agg[d * (long long)D + t], h[s * (long long)D + t]);
    }
}

// ---------------------------------------------------- WMMA fp32 GEMM (fused)
// out[M x 128] = act( (c1*A1 + A2) @ W + bias ),  K = N = 128, fp32 WMMA.
// Block: 256 threads (8 waves). Block tile: 16 rows x 128 cols.
// Wave w owns N-tile w (cols 16w..16w+15); K-loop of 32 x v_wmma_f32_16x16x4_f32.
__global__ __launch_bounds__(256) void gemm_wmma(
    const float* __restrict__ A1, const float* __restrict__ A2,
    const float* __restrict__ epsv, int layer,
    const float* __restrict__ W, const float* __restrict__ bias,
    float* __restrict__ out, int relu) {
    __shared__ float sW[D][132];   // padded stride: conflict-free column reads
    __shared__ float sA[16][132];

    const int tid = threadIdx.x;
    const long long rowBase = (long long)blockIdx.x * 16;
    float c1 = 1.0f;
    if (epsv) c1 = 1.0f + epsv[layer];

    // Stage W (128x128 fp32 = 64 KB) via float4: 16 x 256 lanes * 16B.
    const float4* W4 = (const float4*)W;
    #pragma unroll
    for (int i = 0; i < 16; ++i) {
        int fi = i * 256 + tid;          // float4 index into 128x128
        int r  = (fi * 4) >> 7;
        int c  = (fi * 4) & 127;
        *(float4*)&sW[r][c] = W4[fi];
    }
    // Stage A tile fused:  z = c1*A1 + A2   (A2 == nullptr -> plain A1)
    const float4* A14 = (const float4*)(A1 + rowBase * D);
    const float4* A24 = A2 ? (const float4*)(A2 + rowBase * D) : nullptr;
    #pragma unroll
    for (int i = 0; i < 2; ++i) {
        int fi = i * 256 + tid;
        int r  = (fi * 4) >> 7;
        int c  = (fi * 4) & 127;
        float4 a = A14[fi];
        if (A24) {
            float4 b = A24[fi];
            a.x = c1 * a.x + b.x;  a.y = c1 * a.y + b.y;
            a.z = c1 * a.z + b.z;  a.w = c1 * a.w + b.w;
        }
        *(float4*)&sA[r][c] = a;
    }
    __syncthreads();

    const int lane = tid & 31;
    const int wave = tid >> 5;          // N-tile id 0..7
    const int m    = lane & 15;
    const int half = lane >> 4;         // 0 -> K=0,1 ; 1 -> K=2,3 (ISA A layout)
    const int n    = wave * 16 + m;

    v8f acc = {};
    #pragma unroll
    for (int k = 0; k < D; k += 4) {
        v2f a, b;
        a[0] = sA[m][k + 2 * half + 0];
        a[1] = sA[m][k + 2 * half + 1];
        b[0] = sW[k + 2 * half + 0][n];
        b[1] = sW[k + 2 * half + 1][n];
        acc = __builtin_amdgcn_wmma_f32_16x16x4_f32(
            false, a, false, b, (short)0, acc, false, false);
    }

    // C layout: VGPR r -> rows (r, r+8) split across lane halves; N = lane%16.
    float bn = bias[n];
    #pragma unroll
    for (int r = 0; r < 8; ++r) {
        long long row = rowBase + r + 8 * half;
        float v = acc[r] + bn;
        if (relu) v = fmaxf(v, 0.0f);
        out[row * D + n] = v;
    }
}

// --------------------------------------------------------------- BN statistics
// stats[0..127]=sum, stats[128..255]=sum of squares (global atomics on partials)
__global__ __launch_bounds__(256) void bn_stats(const float* __restrict__ z,
                                                float* __restrict__ stats) {
    __shared__ float ssum[D], ssq[D];
    int t = threadIdx.x;
    if (t < D) { ssum[t] = 0.0f; ssq[t] = 0.0f; }
    __syncthreads();
    int col  = t & 127;
    int r    = blockIdx.x * ROWS_PER_BLOCK + (t >> 7);
    int rend = min(blockIdx.x * ROWS_PER_BLOCK + ROWS_PER_BLOCK, NNODES);
    float ls = 0.0f, lq = 0.0f;
    for (; r < rend; r += 2) {
        float v = z[(long long)r * D + col];
        ls += v; lq += v * v;
    }
    atomicAdd(&ssum[col], ls);
    atomicAdd(&ssq[col], lq);
    __syncthreads();
    if (t < D) {
        atomicAdd(&stats[t], ssum[t]);
        atomicAdd(&stats[D + t], ssq[t]);
    }
}

// --------------------------------- BN apply + ReLU + write h + global_add_pool
__global__ __launch_bounds__(256) void bn_apply_pool(
    const float* __restrict__ z, const float* __restrict__ stats,
    const float* __restrict__ gamma, const float* __restrict__ beta,
    const long long* __restrict__ batch,
    float* __restrict__ h, float* __restrict__ pools, int layer) {
    int t   = threadIdx.x;
    int col = t & 127;
    const float invN = 1.0f / (float)NNODES;
    float mean = stats[col] * invN;
    float var  = stats[D + col] * invN - mean * mean;
    float sc   = gamma[col] * rsqrtf(var + 1e-5f);
    float sh   = beta[col] - mean * sc;
    int r    = blockIdx.x * ROWS_PER_BLOCK + (t >> 7);
    int rend = min(blockIdx.x * ROWS_PER_BLOCK + ROWS_PER_BLOCK, NNODES);
    for (; r < rend; r += 2) {
        float v = fmaxf(z[(long long)r * D + col] * sc + sh, 0.0f);
        h[(long long)r * D + col] = v;
        long long g = batch[r];
        atomicAdd(&pools[g * (NLAYERS * D) + layer * D + col], v);
    }
}

// ------------------------------------------------------------- classifier head
__global__ void final_linear(const float* __restrict__ pools,
                             const float* __restrict__ Wlin,
                             const float* __restrict__ blin,
                             float* __restrict__ out) {
    int g    = blockIdx.x;
    int lane = threadIdx.x;   // one wave32 per graph
    const float* pg = pools + (long long)g * (NLAYERS * D);
    #pragma unroll
    for (int c = 0; c < NCLASSES; ++c) {
        float s = 0.0f;
        for (int k = lane; k < NLAYERS * D; k += 32)
            s += pg[k] * Wlin[k * NCLASSES + c];
        #pragma unroll
        for (int off = 16; off > 0; off >>= 1)
            s += __shfl_down(s, off, 32);
        if (lane == 0) out[g * NCLASSES + c] = s + blin[c];
    }
}

// ----------------------------------------------------------------- entry point
extern "C" void kernel_launch(void* const* d_in, const int* in_sizes, int n_in,
                              void* d_out, int out_size, void* d_ws, size_t ws_size,
                              hipStream_t stream) {
    const float*     x     = (const float*)d_in[0];
    const long long* ei    = (const long long*)d_in[1];
    const long long* src   = ei;
    const long long* dst   = ei + NEDGES;
    const long long* batch = (const long long*)d_in[2];
    const float* W1    = (const float*)d_in[3];
    const float* b1    = (const float*)d_in[4];
    const float* W2    = (const float*)d_in[5];
    const float* b2    = (const float*)d_in[6];
    const float* eps   = (const float*)d_in[7];
    const float* gamma = (const float*)d_in[8];
    const float* beta  = (const float*)d_in[9];
    const float* Wlin  = (const float*)d_in[10];
    const float* blin  = (const float*)d_in[11];
    float* out = (float*)d_out;

    // Workspace: hbuf | agg(also z2) | mid | pools | stats
    float* ws    = (float*)d_ws;
    const size_t NF = (size_t)NNODES * D;          // 6.4M floats
    float* hbuf  = ws;
    float* agg   = ws + NF;
    float* mid   = ws + 2 * NF;
    float* pools = ws + 3 * NF;                    // 512*512
    float* stats = pools + (size_t)NGRAPHS * NLAYERS * D;  // 256

    zero_f32<<<512, 256, 0, stream>>>(pools, NGRAPHS * NLAYERS * D);

    const float* hcur = x;
    for (int l = 0; l < NLAYERS; ++l) {
        zero_f32<<<2048, 256, 0, stream>>>(agg, (int)NF);
        scatter_add_edges<<<NEDGES / 2, 256, 0, stream>>>(hcur, src, dst, agg);
        // GEMM1: relu(((1+eps)*h + agg) @ W1 + b1)  -> mid
        gemm_wmma<<<NNODES / 16, 256, 0, stream>>>(
            hcur, agg, eps, l, W1 + (size_t)l * D * D, b1 + l * D, mid, 1);
        // GEMM2: mid @ W2 + b2 -> z2 (reuse agg buffer)
        gemm_wmma<<<NNODES / 16, 256, 0, stream>>>(
            mid, nullptr, nullptr, l, W2 + (size_t)l * D * D, b2 + l * D, agg, 0);
        zero_f32<<<1, 256, 0, stream>>>(stats, 2 * D);
        bn_stats<<<(NNODES + ROWS_PER_BLOCK - 1) / ROWS_PER_BLOCK, 256, 0, stream>>>(agg, stats);
        bn_apply_pool<<<(NNODES + ROWS_PER_BLOCK - 1) / ROWS_PER_BLOCK, 256, 0, stream>>>(
            agg, stats, gamma + l * D, beta + l * D, batch, hbuf, pools, l);
        hcur = hbuf;
    }
    final_linear<<<NGRAPHS, 32, 0, stream>>>(pools, Wlin, blin, out);
}